// DifferentiableGaussianRandomField_2087354106669
// MI455X (gfx1250) — compile-verified
//
#include <hip/hip_runtime.h>
#include <math.h>

#define NS 128
#define NW 64
#define W_MAX 30.0f
#define L1C 0.2f
#define L2C 0.3f
#define RHOC 0.3f

typedef __attribute__((ext_vector_type(2))) float v2f;
typedef __attribute__((ext_vector_type(8))) float v8f;

// ---------------------------------------------------------------------------
// Stage 1: trig tables + element-wise frequency-grid combinations.
//   Ct[s*64+w] = cos(vs[s]*vw[w]), St = sin(...)            (128x64 each)
// G/H grids are stored TRANSPOSED (indexed [w1][w2]) so that the stage-2
// B-operand fragment {B[k,n], B[k+1,n]} is a contiguous 8-byte load:
//   GcT[w1*64+w2] = A1*cosP1 + A2*cosP2      (feeds Mc via C)
//   GsT[w1*64+w2] = A2*sinP2 - A1*sinP1      (feeds Mc via S)
//   HcT[w1*64+w2] = A1*sinP1 + A2*sinP2      (feeds Ms via C)
//   HsT[w1*64+w2] = A1*cosP1 - A2*cosP2      (feeds Ms via S)
// ---------------------------------------------------------------------------
__global__ void grf_setup_kernel(const float* __restrict__ theta,
                                 float* __restrict__ Ct, float* __restrict__ St,
                                 float* __restrict__ GcT, float* __restrict__ GsT,
                                 float* __restrict__ HcT, float* __restrict__ HsT) {
    const int t = blockIdx.x * blockDim.x + threadIdx.x;   // 0..8191
    const float dw = W_MAX / (float)(NW - 1);

    // trig tables: s = t/64 in [0,128), w = t%64
    {
        const int s = t >> 6, w = t & 63;
        const float x = ((float)s / (float)(NS - 1)) * ((float)w * dw);
        float sn, cs;
        __sincosf(x, &sn, &cs);
        Ct[t] = cs;
        St[t] = sn;
    }

    if (t < NW * NW) {
        const int w1 = t & 63, w2 = t >> 6;
        const float vw1 = (float)w1 * dw;
        const float vw2 = (float)w2 * dw;

        // quadrature mask (matches the reference's .at[] update order)
        float m;
        if (w2 == NW - 1) m = (w1 == 0) ? 0.25f : 1.0f;
        else              m = (w1 == 0) ? 1.0f  : 2.0f;
        m *= dw * dw;

        // anisotropic SE spectral density; A2 uses sdf(w1, -w2) -> cross sign flips
        const float cnorm = (1.0f * L1C * L2C) / (4.0f * (float)M_PI);
        const float x1 = L1C * vw1, x2 = L2C * vw2;
        const float q  = x1 * x1 + x2 * x2;
        const float cr = 2.0f * RHOC * x1 * x2;
        const float A1 = sqrtf(m * cnorm * __expf(-(q + cr) * 0.25f));
        const float A2 = sqrtf(m * cnorm * __expf(-(q - cr) * 0.25f));

        float sp1, cp1, sp2, cp2;
        __sincosf(theta[t],           &sp1, &cp1);
        __sincosf(theta[NW * NW + t], &sp2, &cp2);

        const int tt = w1 * NW + w2;   // transposed store
        GcT[tt] = A1 * cp1 + A2 * cp2;
        GsT[tt] = A2 * sp2 - A1 * sp1;
        HcT[tt] = A1 * sp1 + A2 * sp2;
        HsT[tt] = A1 * cp1 - A2 * cp2;
    }
}

// ---------------------------------------------------------------------------
// Stage 2: Mc = Ct(128x64) @ Gc(64x64) + St @ Gs ; waves 0..31 -> Mc tiles,
// waves 32..63 -> Ms tiles (stored negated so stage 3 is pure accumulation).
// One wave computes one 16x16 f32 tile via V_WMMA_F32_16X16X4_F32, K=64.
// B matrices arrive transposed, so every fragment load is one b64.
// ---------------------------------------------------------------------------
__global__ void grf_gemm1_kernel(const float* __restrict__ Ct, const float* __restrict__ St,
                                 const float* __restrict__ GcT, const float* __restrict__ GsT,
                                 const float* __restrict__ HcT, const float* __restrict__ HsT,
                                 float* __restrict__ Mc, float* __restrict__ Mn) {
    const int lane  = threadIdx.x & 31;
    const int wave  = (blockIdx.x * blockDim.x + threadIdx.x) >> 5;  // 0..63
    const int which = wave >> 5;                                     // 0: Mc, 1: -Ms
    const int tile  = wave & 31;
    const int tm    = tile >> 2;     // 0..7  (rows of 128)
    const int tn    = tile & 3;      // 0..3  (cols of 64)

    const float* B0T = which ? HcT : GcT;
    const float* B1T = which ? HsT : GsT;
    float*       D   = which ? Mn : Mc;

    const int half = lane >> 4;           // 0: K=k0,k0+1 ; 1: K=k0+2,k0+3
    const int row  = tm * 16 + (lane & 15);
    const int col  = tn * 16 + (lane & 15);

    v8f acc = {};
    #pragma unroll
    for (int k0 = 0; k0 < NW; k0 += 4) {
        const int ka = k0 + half * 2;
        const v2f a = *(const v2f*)&Ct [row * NW + ka];
        const v2f b = *(const v2f*)&B0T[col * NW + ka];   // = B0[ka..ka+1, col]
        acc = __builtin_amdgcn_wmma_f32_16x16x4_f32(false, a, false, b,
                                                    (short)0, acc, false, false);
    }
    #pragma unroll
    for (int k0 = 0; k0 < NW; k0 += 4) {
        const int ka = k0 + half * 2;
        const v2f a = *(const v2f*)&St [row * NW + ka];
        const v2f b = *(const v2f*)&B1T[col * NW + ka];
        acc = __builtin_amdgcn_wmma_f32_16x16x4_f32(false, a, false, b,
                                                    (short)0, acc, false, false);
    }

    const float sgn   = which ? -1.0f : 1.0f;
    const int   mbase = tm * 16 + half * 8;   // D layout: VGPR r -> M = mbase + r
    #pragma unroll
    for (int r = 0; r < 8; ++r)
        D[(mbase + r) * NW + col] = sgn * acc[r];
}

// ---------------------------------------------------------------------------
// Stage 3: Out(128x128) = sqrt2 * (Mc @ Ct^T + (-Ms) @ St^T).
// B operand is a transpose of the trig tables, so loads are contiguous b64.
// ---------------------------------------------------------------------------
__global__ void grf_gemm2_kernel(const float* __restrict__ Mc, const float* __restrict__ Mn,
                                 const float* __restrict__ Ct, const float* __restrict__ St,
                                 float* __restrict__ out) {
    const int lane = threadIdx.x & 31;
    const int wave = (blockIdx.x * blockDim.x + threadIdx.x) >> 5;  // 0..63
    const int tm   = wave >> 3;    // 0..7
    const int tn   = wave & 7;     // 0..7

    const int half = lane >> 4;
    const int row  = tm * 16 + (lane & 15);   // s2
    const int col  = tn * 16 + (lane & 15);   // s1

    v8f acc = {};
    #pragma unroll
    for (int k0 = 0; k0 < NW; k0 += 4) {
        const int ka = k0 + half * 2;
        const v2f a = *(const v2f*)&Mc[row * NW + ka];
        const v2f b = *(const v2f*)&Ct[col * NW + ka];    // B[k,n] = Ct[n,k]
        acc = __builtin_amdgcn_wmma_f32_16x16x4_f32(false, a, false, b,
                                                    (short)0, acc, false, false);
    }
    #pragma unroll
    for (int k0 = 0; k0 < NW; k0 += 4) {
        const int ka = k0 + half * 2;
        const v2f a = *(const v2f*)&Mn[row * NW + ka];
        const v2f b = *(const v2f*)&St[col * NW + ka];
        acc = __builtin_amdgcn_wmma_f32_16x16x4_f32(false, a, false, b,
                                                    (short)0, acc, false, false);
    }

    const float SQRT2 = 1.41421356237309505f;
    const int mbase = tm * 16 + half * 8;
    #pragma unroll
    for (int r = 0; r < 8; ++r)
        out[(mbase + r) * NS + col] = SQRT2 * acc[r];
}

// ---------------------------------------------------------------------------
extern "C" void kernel_launch(void* const* d_in, const int* in_sizes, int n_in,
                              void* d_out, int out_size, void* d_ws, size_t ws_size,
                              hipStream_t stream) {
    const float* theta = (const float*)d_in[0];
    float* out = (float*)d_out;

    float* ws  = (float*)d_ws;
    float* Ct  = ws;              // 128*64 = 8192
    float* St  = ws + 8192;       // 8192
    float* GcT = ws + 16384;      // 4096 (transposed)
    float* GsT = ws + 20480;      // 4096
    float* HcT = ws + 24576;      // 4096
    float* HsT = ws + 28672;      // 4096
    float* Mc  = ws + 32768;      // 8192
    float* Mn  = ws + 40960;      // 8192 (= -Ms) -> total 49152 floats = 192 KB

    // Stage 1: 8192 threads (trig tables) incl. first 4096 doing the G/H grids
    grf_setup_kernel<<<32, 256, 0, stream>>>(theta, Ct, St, GcT, GsT, HcT, HsT);
    // Stage 2: 64 waves = 8 blocks x 256 threads (8 waves each)
    grf_gemm1_kernel<<<8, 256, 0, stream>>>(Ct, St, GcT, GsT, HcT, HsT, Mc, Mn);
    // Stage 3: 64 waves
    grf_gemm2_kernel<<<8, 256, 0, stream>>>(Mc, Mn, Ct, St, out);
}